// ALR_51178830299540
// MI455X (gfx1250) — compile-verified
//
#include <hip/hip_runtime.h>
#include <hip/hip_bf16.h>

#define EPSV 1e-8f

typedef float v2f __attribute__((ext_vector_type(2)));
typedef float v8f __attribute__((ext_vector_type(8)));

// ---------------------------------------------------------------------------
// Kernel 1: A0 = 2*(M11 + M22 + R - R^T), M21 = (L L^T)[:64, 64:]
//   M11[i][j] = dot(L_row_i,   L_row_j)    + eps*(i==j)
//   M22[i][j] = dot(L_row_64+i, L_row_64+j) + eps*(i==j)
//   M21[i][j] = dot(L_row_i,   L_row_64+j)
// grid=64 (i), block=64 (j)
// ---------------------------------------------------------------------------
__global__ __launch_bounds__(64) void alr_prep_kernel(
    const float* __restrict__ L, const float* __restrict__ R,
    float* __restrict__ A0, float* __restrict__ M21) {
  const int i = blockIdx.x;
  const int j = threadIdx.x;
  float d11 = 0.f, d22 = 0.f, d21 = 0.f;
  #pragma unroll 4
  for (int k = 0; k < 128; ++k) {
    float li  = L[i * 128 + k];
    float lj  = L[j * 128 + k];
    float li2 = L[(64 + i) * 128 + k];
    float lj2 = L[(64 + j) * 128 + k];
    d11 += li * lj;
    d22 += li2 * lj2;
    d21 += li * lj2;
  }
  const float e = (i == j) ? EPSV : 0.f;
  A0[i * 64 + j]  = 2.f * ((d11 + e) + (d22 + e) + R[i * 64 + j] - R[j * 64 + i]);
  M21[i * 64 + j] = d21;
}

// ---------------------------------------------------------------------------
// Kernel 2: Gauss-Jordan inversion of A0 (64x64) with partial pivoting,
// then Afinal = inv(A0) @ M21.  One block of 64 threads; thread t owns
// columns t and t+64 of the augmented [A0 | I] matrix in LDS.
// ---------------------------------------------------------------------------
__global__ __launch_bounds__(64) void alr_inv_kernel(
    const float* __restrict__ A0, const float* __restrict__ M21,
    float* __restrict__ Afinal) {
  __shared__ float Aug[64][128];
  __shared__ float s_f[64];
  __shared__ int s_pr;
  const int t = threadIdx.x;

  for (int r = 0; r < 64; ++r) {
    Aug[r][t]      = A0[r * 64 + t];
    Aug[r][64 + t] = (r == t) ? 1.f : 0.f;
  }
  __syncthreads();

  for (int p = 0; p < 64; ++p) {
    // pivot search (serial on thread 0; 64 LDS reads, negligible)
    if (t == 0) {
      int pr = p;
      float best = fabsf(Aug[p][p]);
      for (int r = p + 1; r < 64; ++r) {
        float v = fabsf(Aug[r][p]);
        if (v > best) { best = v; pr = r; }
      }
      s_pr = pr;
    }
    __syncthreads();
    const int pr = s_pr;
    if (pr != p) {
      float a  = Aug[p][t],      b  = Aug[pr][t];
      float a2 = Aug[p][64 + t], b2 = Aug[pr][64 + t];
      Aug[p][t] = b;       Aug[pr][t] = a;
      Aug[p][64 + t] = b2; Aug[pr][64 + t] = a2;
    }
    __syncthreads();
    const float pv = Aug[p][p];     // all read pivot before it is scaled
    s_f[t] = Aug[t][p];             // capture elimination factors (pre-scale)
    __syncthreads();
    const float inv = 1.f / pv;
    Aug[p][t]      *= inv;
    Aug[p][64 + t] *= inv;
    __syncthreads();
    const float prow  = Aug[p][t];
    const float prow2 = Aug[p][64 + t];
    for (int r = 0; r < 64; ++r) {
      if (r == p) continue;
      const float f = s_f[r];
      Aug[r][t]      -= f * prow;
      Aug[r][64 + t] -= f * prow2;
    }
    __syncthreads();
  }

  // Afinal = Ainv @ M21 ; thread t computes column t.
  for (int i = 0; i < 64; ++i) {
    float acc = 0.f;
    for (int k = 0; k < 64; ++k)
      acc += Aug[i][64 + k] * M21[k * 64 + t];   // Aug read is lane-uniform (broadcast)
    Afinal[i * 64 + t] = acc;
  }
}

// ---------------------------------------------------------------------------
// Kernel 3: out(64 x NC) = A(64x64) @ X(64 x NC) using V_WMMA_F32_16X16X4_F32.
// One wave per 16-column strip; 4 row-tiles x 16 k-steps = 64 WMMAs per wave.
// A fragments (16x4): lane holds M=lane&15, v0:K=4*kt+2*(lane>=16), v1:K+1
//   -> contiguous in memory -> one aligned b64 load per fragment.
// B fragments (4x16):  lane holds N=j0+(lane&15), v0:K=4*kt+2*(lane>=16), v1:K+1
// C/D (16x16): lanes 0-15 -> M=v, lanes 16-31 -> M=8+v, N=j0+(lane&15).
// ---------------------------------------------------------------------------
__global__ __launch_bounds__(256) void alr_gemm_kernel(
    const float* __restrict__ X, const float* __restrict__ A,
    float* __restrict__ out, int NC) {
  const int lane  = threadIdx.x & 31;
  const int wave  = threadIdx.x >> 5;
  const int tile  = blockIdx.x * 8 + wave;
  const int j0    = tile * 16;
  if (j0 >= NC) return;                 // wave-uniform: EXEC stays all-ones
  const int mrow  = lane & 15;
  const int khalf = lane >> 4;          // 0 or 1
  const int col   = j0 + mrow;

  // Load all A fragments (A stays resident: 128 VGPRs)
  v2f afrag[4][16];
  #pragma unroll
  for (int mt = 0; mt < 4; ++mt) {
    #pragma unroll
    for (int kt = 0; kt < 16; ++kt) {
      const float* p = A + (mt * 16 + mrow) * 64 + kt * 4 + 2 * khalf;
      afrag[mt][kt] = *(const v2f*)p;   // 8-byte aligned b64 load
    }
  }

  // Preload all B fragments for this 16-column strip (32 loads in flight)
  v2f bfrag[16];
  #pragma unroll
  for (int kt = 0; kt < 16; ++kt) {
    const int k = kt * 4 + 2 * khalf;
    v2f b;
    b.x = X[k * NC + col];
    b.y = X[(k + 1) * NC + col];
    bfrag[kt] = b;
  }

  v8f c0 = {}, c1 = {}, c2 = {}, c3 = {};
  #pragma unroll
  for (int kt = 0; kt < 16; ++kt) {
    c0 = __builtin_amdgcn_wmma_f32_16x16x4_f32(false, afrag[0][kt], false, bfrag[kt],
                                               (short)0, c0, false, false);
    c1 = __builtin_amdgcn_wmma_f32_16x16x4_f32(false, afrag[1][kt], false, bfrag[kt],
                                               (short)0, c1, false, false);
    c2 = __builtin_amdgcn_wmma_f32_16x16x4_f32(false, afrag[2][kt], false, bfrag[kt],
                                               (short)0, c2, false, false);
    c3 = __builtin_amdgcn_wmma_f32_16x16x4_f32(false, afrag[3][kt], false, bfrag[kt],
                                               (short)0, c3, false, false);
  }

  // Store D tiles per the 16x16 f32 C/D layout
  #pragma unroll
  for (int r = 0; r < 8; ++r) {
    const int ib = r + 8 * khalf;
    out[(ib +  0) * NC + col] = c0[r];
    out[(ib + 16) * NC + col] = c1[r];
    out[(ib + 32) * NC + col] = c2[r];
    out[(ib + 48) * NC + col] = c3[r];
  }
}

// ---------------------------------------------------------------------------
extern "C" void kernel_launch(void* const* d_in, const int* in_sizes, int n_in,
                              void* d_out, int out_size, void* d_ws, size_t ws_size,
                              hipStream_t stream) {
  const float* x = (const float*)d_in[0];   // (524288, 64) fp32, flat
  const float* L = (const float*)d_in[1];   // (128, 128) fp32
  const float* R = (const float*)d_in[2];   // (64, 64) fp32
  float* out = (float*)d_out;               // (524288, 64) fp32 == (64 x NC) flat
  float* ws  = (float*)d_ws;

  float* A0  = ws;          // 64*64
  float* M21 = ws + 4096;   // 64*64
  float* Af  = ws + 8192;   // 64*64 final A = inv(A0) @ M21

  const int NC = in_sizes[0] / 64;          // 524288 columns

  alr_prep_kernel<<<64, 64, 0, stream>>>(L, R, A0, M21);
  alr_inv_kernel<<<1, 64, 0, stream>>>(A0, M21, Af);

  const int tiles  = NC / 16;               // 16 columns per wave
  const int blocks = (tiles + 7) / 8;       // 8 waves per 256-thread block
  alr_gemm_kernel<<<blocks, 256, 0, stream>>>(x, Af, out, NC);
}